// NodeClassificationHead_38792144618153
// MI455X (gfx1250) — compile-verified
//
#include <hip/hip_runtime.h>
#include <hip/hip_bf16.h>

typedef float v2f __attribute__((ext_vector_type(2)));
typedef float v8f __attribute__((ext_vector_type(8)));

#define F_IN  256
#define F_OUT 32

static inline int cdiv_host(long long a, long long b) { return (int)((a + b - 1) / b); }

// ---------------------------------------------------------------------------
// Zero a word region (d_out / degree scratch are poisoned by the harness).
// ---------------------------------------------------------------------------
__global__ void zero_words_kernel(unsigned* __restrict__ p, long long n) {
    long long i = (long long)blockIdx.x * blockDim.x + threadIdx.x;
    long long stride = (long long)gridDim.x * blockDim.x;
    for (; i < n; i += stride) p[i] = 0u;
}

// ---------------------------------------------------------------------------
// Degree counting: integer atomics (exact, deterministic).
// ---------------------------------------------------------------------------
__global__ void degree_kernel(const int* __restrict__ src, const int* __restrict__ dst,
                              unsigned* __restrict__ deg_src, unsigned* __restrict__ deg_dst,
                              int E) {
    int e = blockIdx.x * blockDim.x + threadIdx.x;
    if (e < E) {
        atomicAdd(&deg_src[src[e]], 1u);
        atomicAdd(&deg_dst[dst[e]], 1u);
    }
}

// ---------------------------------------------------------------------------
// h = (X * rsqrt(max(out_deg,1))) @ W  via V_WMMA_F32_16X16X4_F32.
// One wave computes a 16x32 output tile: two 16x16 f32 accumulators.
// A tile (16x256 f32) staged in LDS with coalesced loads; row stride padded
// to 260 floats so the A-fragment reads (16 rows, same column) hit 16
// distinct banks (bank = (4*m + col) % 64).
//
// 32-bit A 16x4 layout: lanes 0-15: M=lane,   v0=K+0, v1=K+1
//                       lanes16-31: M=lane-16,v0=K+2, v1=K+3
// 32-bit B 4x16 layout (mirror): lanes 0-15: N=lane, v0=K+0, v1=K+1
//                                lanes16-31: N=lane-16, v0=K+2, v1=K+3
// 32-bit D 16x16 layout: VGPR g: lanes 0-15: M=g, N=lane;
//                                lanes16-31: M=8+g, N=lane-16
// ---------------------------------------------------------------------------
#define GEMM_WAVES 2
__global__ __launch_bounds__(GEMM_WAVES * 32)
void gemm_wmma_kernel(const float* __restrict__ x, const float* __restrict__ W,
                      const unsigned* __restrict__ deg_src,
                      float* __restrict__ h, int N) {
    __shared__ float xs[GEMM_WAVES][16][F_IN + 4];

    const int lane = threadIdx.x & 31;
    const int wave = threadIdx.x >> 5;
    const int tile = blockIdx.x * GEMM_WAVES + wave;
    const int row0 = tile * 16;

    // Stage 16x256 tile into LDS, fully coalesced (128 floats per lane).
    // Tail tiles clamp the row; results are masked at store time.
    for (int i = lane; i < 16 * F_IN; i += 32) {
        int r = i >> 8;            // / F_IN
        int c = i & (F_IN - 1);
        int rr = row0 + r; if (rr > N - 1) rr = N - 1;
        xs[wave][r][c] = x[(long long)rr * F_IN + c];
    }
    __syncthreads();

    const int m     = lane & 15;            // A row / B column within half
    const int khalf = (lane >> 4) << 1;     // 0 for lanes 0-15, 2 for 16-31

    int nrow = row0 + m; if (nrow > N - 1) nrow = N - 1;
    const float nrm = rsqrtf(fmaxf((float)deg_src[nrow], 1.0f));

    v8f c0 = {};   // output columns 0..15
    v8f c1 = {};   // output columns 16..31

#pragma unroll 8
    for (int k = 0; k < F_IN; k += 4) {
        v2f a;
        a.x = xs[wave][m][k + khalf]     * nrm;
        a.y = xs[wave][m][k + khalf + 1] * nrm;

        const float* Wk = W + (long long)(k + khalf) * F_OUT;
        v2f b0, b1;
        b0.x = Wk[m];
        b0.y = Wk[F_OUT + m];
        b1.x = Wk[16 + m];
        b1.y = Wk[F_OUT + 16 + m];

        c0 = __builtin_amdgcn_wmma_f32_16x16x4_f32(false, a, false, b0,
                                                   (short)0, c0, false, false);
        c1 = __builtin_amdgcn_wmma_f32_16x16x4_f32(false, a, false, b1,
                                                   (short)0, c1, false, false);
    }

    // Store D: VGPR g -> row (lane<16 ? g : 8+g), col (lane&15) (+16 for c1)
    const int mbase = (lane >> 4) * 8;
#pragma unroll
    for (int g = 0; g < 8; ++g) {
        int r = row0 + mbase + g;
        if (r < N) {
            h[(long long)r * F_OUT + m]      = c0[g];
            h[(long long)r * F_OUT + 16 + m] = c1[g];
        }
    }
}

// ---------------------------------------------------------------------------
// Edge scatter: one wave per edge, lane = feature (F_OUT==32==wave32).
// h and out both fit in the 192MB L2, so this runs at L2 bandwidth.
// ---------------------------------------------------------------------------
__global__ __launch_bounds__(256)
void scatter_kernel(const int* __restrict__ src, const int* __restrict__ dst,
                    const float* __restrict__ h, float* __restrict__ out, int E) {
    long long gtid = (long long)blockIdx.x * blockDim.x + threadIdx.x;
    int e    = (int)(gtid >> 5);
    int lane = threadIdx.x & 31;
    if (e < E) {
        int s = src[e];            // uniform across wave -> one cacheline
        int d = dst[e];
        float v = h[(long long)s * F_OUT + lane];
        atomicAdd(&out[(long long)d * F_OUT + lane], v);
    }
}

// ---------------------------------------------------------------------------
// out = out * rsqrt(max(in_deg,1)) + b
// ---------------------------------------------------------------------------
__global__ void finalize_kernel(float* __restrict__ out, const unsigned* __restrict__ deg_dst,
                                const float* __restrict__ b, int N) {
    long long i = (long long)blockIdx.x * blockDim.x + threadIdx.x;
    if (i < (long long)N * F_OUT) {
        int n = (int)(i >> 5);
        int f = (int)(i & 31);
        float nd = rsqrtf(fmaxf((float)deg_dst[n], 1.0f));
        out[i] = out[i] * nd + b[f];
    }
}

// ---------------------------------------------------------------------------
// d_in: 0=in_feat [N,256] f32, 1=W [256,32] f32, 2=b [32] f32,
//       3=src [E] i32, 4=dst [E] i32.   d_out: [N,32] f32.
// ws layout: deg_src (N u32) | deg_dst (N u32) | h (N*32 f32)
// ---------------------------------------------------------------------------
extern "C" void kernel_launch(void* const* d_in, const int* in_sizes, int n_in,
                              void* d_out, int out_size, void* d_ws, size_t ws_size,
                              hipStream_t stream) {
    const float* x   = (const float*)d_in[0];
    const float* W   = (const float*)d_in[1];
    const float* b   = (const float*)d_in[2];
    const int*   src = (const int*)d_in[3];
    const int*   dst = (const int*)d_in[4];
    float* out = (float*)d_out;

    const int N = in_sizes[0] / F_IN;
    const int E = in_sizes[3];

    unsigned* deg_src = (unsigned*)d_ws;
    unsigned* deg_dst = deg_src + N;
    float*    h       = (float*)(deg_dst + N);

    // Zero degree scratch and the (poisoned) output accumulator.
    zero_words_kernel<<<cdiv_host(2LL * N, 256), 256, 0, stream>>>((unsigned*)d_ws, 2LL * N);
    zero_words_kernel<<<cdiv_host((long long)N * F_OUT, 256), 256, 0, stream>>>(
        (unsigned*)d_out, (long long)N * F_OUT);

    degree_kernel<<<cdiv_host(E, 256), 256, 0, stream>>>(src, dst, deg_src, deg_dst, E);

    const int tiles = cdiv_host(N, 16);
    gemm_wmma_kernel<<<cdiv_host(tiles, GEMM_WAVES), GEMM_WAVES * 32, 0, stream>>>(
        x, W, deg_src, h, N);

    scatter_kernel<<<cdiv_host((long long)E * 32, 256), 256, 0, stream>>>(src, dst, h, out, E);

    finalize_kernel<<<cdiv_host((long long)N * F_OUT, 256), 256, 0, stream>>>(
        out, deg_dst, b, N);
}